// LossFunction_11003706213253
// MI455X (gfx1250) — compile-verified
//
#include <hip/hip_runtime.h>
#include <stdint.h>

#define N_ 20000
#define T_ 512
#define K_ 12
#define F_ 8
#define WPB (T_ / 32)   // waves per block = 16

__device__ __forceinline__ float waveReduceSum(float v) {
    #pragma unroll
    for (int m = 16; m > 0; m >>= 1) v += __shfl_xor(v, m, 32);
    return v;
}

__global__ __launch_bounds__(T_) void garch_row_kernel(
    const float* __restrict__ y,
    const float* __restrict__ f,
    const float* __restrict__ theta,
    float* __restrict__ out_eps,
    float* __restrict__ out_sigma2,
    float* __restrict__ partial_loss)
{
    __shared__ __align__(16) float sTheta[T_ * K_];   // 24576 B: theta[n,1:,:]
    __shared__ float sEps2[T_];
    __shared__ float sAggA[WPB];
    __shared__ float sAggC[WPB];
    __shared__ float sRedW[WPB];
    __shared__ float sRedA[WPB];
    __shared__ float sS0;

    const int n    = blockIdx.x;
    const int t    = threadIdx.x;
    const int lane = t & 31;
    const int wid  = t >> 5;

    // ---- Issue async global->LDS copies for theta[n, 1:, :] (48 B per t).
    // INST_OFFSET applies to both global and LDS addresses, so one address
    // pair covers all three 16B chunks.
    {
        const float* thRow = theta + ((size_t)n * (T_ + 1) + 1) * K_;
        uint64_t g = (uint64_t)(uintptr_t)thRow + (uint64_t)t * 48u;
        uint32_t l = (uint32_t)(uintptr_t)sTheta + (uint32_t)t * 48u;
        asm volatile(
            "global_load_async_to_lds_b128 %0, %1, off\n\t"
            "global_load_async_to_lds_b128 %0, %1, off offset:16\n\t"
            "global_load_async_to_lds_b128 %0, %1, off offset:32"
            :: "v"(l), "v"(g) : "memory");
    }

    // ---- Overlap: fetch y and f through the normal VMEM path while the
    // async engine fills LDS (scheduled between the two asm blocks).
    float yv = y[(size_t)n * T_ + t];
    const float4* fv = (const float4*)(f + (size_t)t * F_);
    float4 f0 = fv[0], f1 = fv[1];

    asm volatile("s_wait_asynccnt 0" ::: "memory");
    __syncthreads();   // all waves' async copies landed in LDS

    // ---- eps / eps2 / per-t params
    const float4* thv = (const float4*)(&sTheta[t * K_]);
    float4 p0 = thv[0];            // th[0..3]
    float4 p1 = thv[1];            // th[4..7]
    float4 p2 = thv[2];            // th[8..11]

    float dot = p0.y * f0.x + p0.z * f0.y + p0.w * f0.z + p1.x * f0.w
              + p1.y * f1.x + p1.z * f1.y + p1.w * f1.z + p2.x * f1.w;
    float eps  = yv - p0.x - dot;
    float a = p2.y, b = p2.z, w = p2.w;
    float apb  = a + b;
    float eps2 = eps * eps;

    out_eps[(size_t)n * T_ + t] = eps;
    sEps2[t] = eps2;

    // row reductions for sigma2_0
    float rw = waveReduceSum(w);
    float ra = waveReduceSum(apb);
    if (lane == 0) { sRedW[wid] = rw; sRedA[wid] = ra; }
    __syncthreads();

    if (t == 0) {
        float sw = 0.f, sa = 0.f;
        #pragma unroll
        for (int i = 0; i < WPB; ++i) { sw += sRedW[i]; sa += sRedA[i]; }
        float mw = sw * (1.0f / (float)T_);
        float ma = sa * (1.0f / (float)T_);
        sS0 = w + apb * (mw / (1.0f - ma));   // thread 0 holds t==0 values
    }

    // ---- affine-pair scan: s_t = a_t*s_{t-1} + c_t, c_t = w_t + b_t*eps2_{t-1}
    float pa, pc;
    if (t == 0) { pa = 1.0f; pc = 0.0f; }                 // identity
    else        { pa = a;    pc = w + b * sEps2[t - 1]; }

    #pragma unroll
    for (int off = 1; off < 32; off <<= 1) {
        float ua = __shfl_up(pa, off, 32);
        float uc = __shfl_up(pc, off, 32);
        if (lane >= off) { pc = pa * uc + pc; pa = pa * ua; }
    }
    if (lane == 31) { sAggA[wid] = pa; sAggC[wid] = pc; }
    __syncthreads();

    // exclusive prefix of wave aggregates (<=15 serial composes)
    float Ea = 1.0f, Ec = 0.0f;
    for (int i = 0; i < wid; ++i) {
        Ec = sAggA[i] * Ec + sAggC[i];
        Ea = sAggA[i] * Ea;
    }
    float A = pa * Ea;
    float C = pa * Ec + pc;

    float sigma2 = A * sS0 + C;
    out_sigma2[(size_t)n * T_ + t] = sigma2;

    // ---- loss contribution (deterministic block reduce -> per-row partial)
    float v  = eps2 / sigma2 + logf(sigma2);
    float rv = waveReduceSum(v);
    __syncthreads();                       // safe to reuse sRedW now
    if (lane == 0) sRedW[wid] = rv;
    __syncthreads();
    if (t == 0) {
        float s = 0.f;
        #pragma unroll
        for (int i = 0; i < WPB; ++i) s += sRedW[i];
        partial_loss[n] = s;
    }
}

__global__ __launch_bounds__(1024) void loss_reduce_kernel(
    const float* __restrict__ partial, float* __restrict__ out_loss)
{
    __shared__ float sRed[32];
    float acc = 0.f;
    for (int i = threadIdx.x; i < N_; i += blockDim.x) acc += partial[i];
    acc = waveReduceSum(acc);
    int lane = threadIdx.x & 31, wid = threadIdx.x >> 5;
    if (lane == 0) sRed[wid] = acc;
    __syncthreads();
    if (wid == 0) {
        float v = sRed[lane];
        v = waveReduceSum(v);
        if (lane == 0) out_loss[0] = v * (1.0f / ((float)N_ * (float)T_));
    }
}

extern "C" void kernel_launch(void* const* d_in, const int* in_sizes, int n_in,
                              void* d_out, int out_size, void* d_ws, size_t ws_size,
                              hipStream_t stream) {
    (void)in_sizes; (void)n_in; (void)out_size; (void)ws_size;
    const float* y     = (const float*)d_in[0];
    const float* f     = (const float*)d_in[1];
    const float* theta = (const float*)d_in[2];

    float* out        = (float*)d_out;
    float* out_loss   = out;                         // [0]
    float* out_eps    = out + 1;                     // [1 .. N*T]
    float* out_sigma2 = out + 1 + (size_t)N_ * T_;   // [1+N*T .. 1+2*N*T)
    float* partial    = (float*)d_ws;                // N_ floats of scratch

    garch_row_kernel<<<N_, T_, 0, stream>>>(y, f, theta, out_eps, out_sigma2, partial);
    loss_reduce_kernel<<<1, 1024, 0, stream>>>(partial, out_loss);
}